// MatrixModel_12884901888369
// MI455X (gfx1250) — compile-verified
//
#include <hip/hip_runtime.h>
#include <hip/hip_bf16.h>

// IPF / Sinkhorn on A=|W| collapses to diagonal scalings:
//   u <- 1/(A v) ; v <- 1/(A^T u) ; out = diag(u) A diag(v)
// 2 matrix read-passes per iteration + 1 read/write pass at the end:
// ~5.6 GB total -> ~240us at 23.3 TB/s HBM. Compute is negligible, so the
// col pass uses V_WMMA_F32_16X16X4_F32 to do the K-reduction in the
// accumulator (no shuffles), with the matrix tile in the B operand
// (coalesced) and u broadcast in the A operand.

#define IPF_N 8192
#define IPF_ITERS 10
#define COL_CHUNKS 32            // row-chunks for the column pass (256 rows each)
#define CHUNK_ROWS (IPF_N / COL_CHUNKS)

typedef __attribute__((ext_vector_type(2))) float v2f;
typedef __attribute__((ext_vector_type(8))) float v8f;

__global__ void ipf_init_v(float* __restrict__ v) {
    int j = blockIdx.x * blockDim.x + threadIdx.x;
    if (j < IPF_N) v[j] = 1.0f;
}

// u[i] = 1 / sum_j |W[i][j]| * v[j]   (one block per row, coalesced float4 stream)
__global__ void ipf_rowpass(const float* __restrict__ W,
                            const float* __restrict__ v,
                            float* __restrict__ u) {
    const int row = blockIdx.x;
    const float4* wr = reinterpret_cast<const float4*>(W + (size_t)row * IPF_N);
    const float4* v4 = reinterpret_cast<const float4*>(v);
    float acc = 0.0f;
    for (int j = threadIdx.x; j < IPF_N / 4; j += blockDim.x) {
        float4 w = wr[j];
        float4 vv = v4[j];
        acc += fabsf(w.x) * vv.x + fabsf(w.y) * vv.y +
               fabsf(w.z) * vv.z + fabsf(w.w) * vv.w;
    }
    __shared__ float red[256];
    red[threadIdx.x] = acc;
    __syncthreads();
    for (int s = 128; s > 0; s >>= 1) {
        if (threadIdx.x < s) red[threadIdx.x] += red[threadIdx.x + s];
        __syncthreads();
    }
    if (threadIdx.x == 0) u[row] = 1.0f / red[0];
}

// Column-sum pass via WMMA: partial[chunk][j] = sum_{i in chunk} u[i]*|W[i][j]|
// Block = 256 threads = 8 waves; each wave owns 16 columns; block covers 128
// columns (grid.x = 64) over one 256-row chunk (grid.y = 32).
// A-fragment (16x4 f32): all 16 rows = u[k..k+3]  (broadcast from LDS)
// B-fragment (4x16  f32): the |W| tile, lane-striped along columns (coalesced)
// D[m][n] accumulates sum_k u[k]*|W[k][j0+n]|; all rows of D identical, and
// lane n's c[0] is exactly the column partial for j0 + (lane & 15).
__global__ void ipf_colpass_wmma(const float* __restrict__ W,
                                 const float* __restrict__ u,
                                 float* __restrict__ partial) {
    const int lane = threadIdx.x & 31;
    const int wave = threadIdx.x >> 5;
    const int half = lane >> 4;          // 0: lanes 0-15, 1: lanes 16-31
    const int nn   = lane & 15;          // column within the wave's 16-col strip
    const int r0   = blockIdx.y * CHUNK_ROWS;
    const int j0   = blockIdx.x * 128 + wave * 16;

    __shared__ float lds_u[CHUNK_ROWS];
    lds_u[threadIdx.x] = u[r0 + threadIdx.x];   // blockDim.x == CHUNK_ROWS == 256
    __syncthreads();

    v8f c = {};
    const size_t colbase = (size_t)j0 + nn;
    for (int kk = 0; kk < CHUNK_ROWS; kk += 4) {
        // B operand: rows (kk+2*half) and (kk+2*half+1) of the 4x16 tile
        const float* rp = W + (size_t)(r0 + kk + 2 * half) * IPF_N + colbase;
        v2f b;
        b.x = fabsf(rp[0]);
        b.y = fabsf(rp[IPF_N]);
        // A operand: broadcast u for K = {0,2}+half*? -> K=2*half and K=2*half+1
        v2f a;
        a.x = lds_u[kk + 2 * half];
        a.y = lds_u[kk + 2 * half + 1];
        // 8 args: (neg_a, A, neg_b, B, c_mod, C, reuse_a, reuse_b)
        c = __builtin_amdgcn_wmma_f32_16x16x4_f32(
            false, a, false, b, (short)0, c, false, false);
    }
    // c[0] on every lane holds the column partial for column j0 + (lane&15)
    if (lane < 16) {
        partial[(size_t)blockIdx.y * IPF_N + j0 + lane] = c[0];
    }
}

// v[j] = 1 / sum_chunks partial[chunk][j]
__global__ void ipf_colreduce(const float* __restrict__ partial,
                              float* __restrict__ v) {
    int j = blockIdx.x * blockDim.x + threadIdx.x;
    float t = 0.0f;
    for (int cidx = 0; cidx < COL_CHUNKS; ++cidx)
        t += partial[(size_t)cidx * IPF_N + j];
    v[j] = 1.0f / t;
}

// out[i][j] = u[i] * |W[i][j]| * v[j]
__global__ void ipf_finalize(const float* __restrict__ W,
                             const float* __restrict__ u,
                             const float* __restrict__ v,
                             float* __restrict__ out) {
    const int row = blockIdx.x;
    const float ui = u[row];
    const float4* wr = reinterpret_cast<const float4*>(W + (size_t)row * IPF_N);
    const float4* v4 = reinterpret_cast<const float4*>(v);
    float4* o4 = reinterpret_cast<float4*>(out + (size_t)row * IPF_N);
    for (int j = threadIdx.x; j < IPF_N / 4; j += blockDim.x) {
        float4 w = wr[j];
        float4 vv = v4[j];
        float4 r;
        r.x = ui * fabsf(w.x) * vv.x;
        r.y = ui * fabsf(w.y) * vv.y;
        r.z = ui * fabsf(w.z) * vv.z;
        r.w = ui * fabsf(w.w) * vv.w;
        o4[j] = r;
    }
}

extern "C" void kernel_launch(void* const* d_in, const int* in_sizes, int n_in,
                              void* d_out, int out_size, void* d_ws, size_t ws_size,
                              hipStream_t stream) {
    (void)in_sizes; (void)n_in; (void)out_size; (void)ws_size;
    const float* W = (const float*)d_in[0];
    float* out = (float*)d_out;

    float* v = (float*)d_ws;                 // 8192 floats
    float* u = v + IPF_N;                    // 8192 floats
    float* partial = u + IPF_N;              // 32 * 8192 floats (1 MB)

    ipf_init_v<<<IPF_N / 256, 256, 0, stream>>>(v);

    for (int it = 0; it < IPF_ITERS; ++it) {
        ipf_rowpass<<<IPF_N, 256, 0, stream>>>(W, v, u);
        ipf_colpass_wmma<<<dim3(IPF_N / 128, COL_CHUNKS), 256, 0, stream>>>(W, u, partial);
        ipf_colreduce<<<IPF_N / 256, 256, 0, stream>>>(partial, v);
    }

    ipf_finalize<<<IPF_N, 256, 0, stream>>>(W, u, v, out);
}